// LinkPredictor_72499047956831
// MI455X (gfx1250) — compile-verified
//
#include <hip/hip_runtime.h>
#include <stddef.h>

typedef __attribute__((ext_vector_type(2))) float v2f;
typedef __attribute__((ext_vector_type(8))) float v8f;

__device__ __forceinline__ v8f wmma4(v2f a, v2f b, v8f c) {
  return __builtin_amdgcn_wmma_f32_16x16x4_f32(
      /*neg_a=*/false, a, /*neg_b=*/false, b,
      /*c_mod=*/(short)0, c, /*reuse_a=*/false, /*reuse_b=*/false);
}

// ---------------------------------------------------------------------------
// WMMA f32 GEMM: C[M,N] = A[M,K] * B[K,N].  N,K compile-time -> all loads get
// immediate offsets; fully unrolled K loop; one wave per 16(M)x64(N) C strip
// (4 accumulators, A-frag reused 4x, 4 independent WMMAs per K-step).
// blockDim = 32 (one wave); grid = (M/16, N/64).
// A frag (16x4 f32): lanes 0-15 M=0..15 {K,K+1}; lanes 16-31 M=0..15 {K+2,K+3}
// B frag (4x16 f32): lanes 0-15 rows {K,K+1}; lanes 16-31 rows {K+2,K+3}
// C (16x16 f32):     vgpr i: lanes 0-15 -> M=i, lanes 16-31 -> M=i+8
// ---------------------------------------------------------------------------
template <int N, int K>
__global__ void gemm_wmma_f32(const float* __restrict__ A,
                              const float* __restrict__ B,
                              float* __restrict__ C) {
  const int lane = threadIdx.x & 31;
  const int m0 = blockIdx.x * 16;
  const int n0 = blockIdx.y * 64;
  const int half = lane >> 4;      // 0: lanes 0-15, 1: lanes 16-31
  const int r    = lane & 15;      // row (A) / col (B,C)

  const float* Arow = A + (size_t)(m0 + r) * K + 2 * half;
  const float* Bcol = B + (size_t)(2 * half) * N + n0 + r;

  v8f c0 = {}, c1 = {}, c2 = {}, c3 = {};
#pragma unroll
  for (int k = 0; k < K; k += 4) {
    v2f a = *(const v2f*)(Arow + k);        // global_load_b64, imm offset
    v2f b0, b1, b2, b3;                     // 8x global_load_b32, imm offsets
    b0.x = Bcol[k * N + 0];       b0.y = Bcol[k * N + N + 0];
    b1.x = Bcol[k * N + 16];      b1.y = Bcol[k * N + N + 16];
    b2.x = Bcol[k * N + 32];      b2.y = Bcol[k * N + N + 32];
    b3.x = Bcol[k * N + 48];      b3.y = Bcol[k * N + N + 48];
    c0 = wmma4(a, b0, c0);
    c1 = wmma4(a, b1, c1);
    c2 = wmma4(a, b2, c2);
    c3 = wmma4(a, b3, c3);
  }

  float* Crow = C + (size_t)(m0 + 8 * half) * N + n0 + r;
#pragma unroll
  for (int i = 0; i < 8; ++i) {
    Crow[i * N + 0]  = c0[i];
    Crow[i * N + 16] = c1[i];
    Crow[i * N + 32] = c2[i];
    Crow[i * N + 48] = c3[i];
  }
}

// ---------------------------------------------------------------------------
// Fold attention vector into weight matrix: Wa[k,h] = sum_f W[k, h*F+f]*att[h,f]
// ---------------------------------------------------------------------------
template <int H, int F>
__global__ void make_wa(const float* __restrict__ W, const float* __restrict__ att,
                        float* __restrict__ Wa, int K) {
  int t = blockIdx.x * blockDim.x + threadIdx.x;
  if (t >= K * H) return;
  int k = t / H, h = t & (H - 1);
  const float* wr = W + (size_t)k * (H * F) + h * F;
  const float* ar = att + h * F;
  float acc = 0.f;
#pragma unroll 4
  for (int f = 0; f < F; ++f) acc += wr[f] * ar[f];
  Wa[t] = acc;
}

// a[n,h] = sum_k X[n,k] * Wa[k,h]
template <int K, int H>
__global__ void proj_att(const float* __restrict__ X, const float* __restrict__ Wa,
                         float* __restrict__ out, int N) {
  int t = blockIdx.x * blockDim.x + threadIdx.x;
  if (t >= N * H) return;
  int n = t / H, h = t & (H - 1);
  const float* xr = X + (size_t)n * K;
  float acc = 0.f;
#pragma unroll 4
  for (int k = 0; k < K; ++k) acc += xr[k] * Wa[k * H + h];
  out[t] = acc;
}

// ---------------------------------------------------------------------------
// Order-preserving float <-> uint key for atomicMax-based segment max
// ---------------------------------------------------------------------------
__device__ __forceinline__ unsigned fkey(float f) {
  unsigned u = __float_as_uint(f);
  return (u & 0x80000000u) ? ~u : (u | 0x80000000u);
}
__device__ __forceinline__ float fdecode(unsigned k) {
  unsigned u = (k & 0x80000000u) ? (k & 0x7FFFFFFFu) : ~k;
  return __uint_as_float(u);
}
__device__ __forceinline__ float leaky(float a) {
  return a > 0.f ? a : 0.2f * a;
}

// Pass 1: alpha = leaky_relu(a_s[src]+a_d[dst]); segment max over dst
template <int H>
__global__ void edge_max(const int* __restrict__ src, const int* __restrict__ dst,
                         const float* __restrict__ as, const float* __restrict__ ad,
                         unsigned* __restrict__ amax, int E) {
  int t = blockIdx.x * blockDim.x + threadIdx.x;
  if (t >= E * H) return;
  int e = t / H, h = t & (H - 1);
  float a = leaky(as[src[e] * H + h] + ad[dst[e] * H + h]);
  atomicMax(&amax[dst[e] * H + h], fkey(a));
}

// Pass 2: denom[dst,h] += exp(alpha - amax[dst,h])
template <int H>
__global__ void edge_denom(const int* __restrict__ src, const int* __restrict__ dst,
                           const float* __restrict__ as, const float* __restrict__ ad,
                           const unsigned* __restrict__ amax, float* __restrict__ den,
                           int E) {
  int t = blockIdx.x * blockDim.x + threadIdx.x;
  if (t >= E * H) return;
  int e = t / H, h = t & (H - 1);
  int d = dst[e] * H + h;
  float a = leaky(as[src[e] * H + h] + ad[d]);
  float ex = __expf(a - fdecode(amax[d]));
  atomicAdd(&den[d], ex);
}

// Pass 3: out[dst, :] += w * xs[src, :]; one wave per edge, HF = H<<LOGF feats
template <int H, int LOGF>
__global__ void edge_aggregate(const int* __restrict__ src, const int* __restrict__ dst,
                               const float* __restrict__ as, const float* __restrict__ ad,
                               const unsigned* __restrict__ amax,
                               const float* __restrict__ den,
                               const float* __restrict__ xs, float* __restrict__ out,
                               int E) {
  int e = blockIdx.x * (blockDim.x >> 5) + (threadIdx.x >> 5);
  if (e >= E) return;
  int lane = threadIdx.x & 31;
  int s = src[e], d = dst[e];
  constexpr int HF = H << LOGF;
#pragma unroll
  for (int p = lane; p < HF; p += 32) {
    int h = p >> LOGF;
    float a = leaky(as[s * H + h] + ad[d * H + h]);
    float ex = __expf(a - fdecode(amax[d * H + h]));
    float w = ex / (den[d * H + h] + 1e-16f);
    atomicAdd(&out[(size_t)d * HF + p], w * xs[(size_t)s * HF + p]);
  }
}

__global__ void bias_relu(float* __restrict__ h, const float* __restrict__ b,
                          int total, int HF) {
  int t = blockIdx.x * blockDim.x + threadIdx.x;
  if (t >= total) return;
  float v = h[t] + b[t & (HF - 1)];
  h[t] = v > 0.f ? v : 0.f;
}

__global__ void bias_add(float* __restrict__ o, const float* __restrict__ b,
                         int total, int C) {
  int t = blockIdx.x * blockDim.x + threadIdx.x;
  if (t >= total) return;
  o[t] = o[t] + b[t & (C - 1)];
}

extern "C" void kernel_launch(void* const* d_in, const int* in_sizes, int n_in,
                              void* d_out, int out_size, void* d_ws, size_t ws_size,
                              hipStream_t stream) {
  (void)n_in; (void)out_size; (void)ws_size;
  const float* x     = (const float*)d_in[0];
  const int*   ei    = (const int*)d_in[1];
  const float* W1s   = (const float*)d_in[2];
  const float* W1d   = (const float*)d_in[3];
  const float* att1s = (const float*)d_in[4];
  const float* att1d = (const float*)d_in[5];
  const float* b1    = (const float*)d_in[6];
  const float* W2s   = (const float*)d_in[7];
  const float* W2d   = (const float*)d_in[8];
  const float* att2s = (const float*)d_in[9];
  const float* att2d = (const float*)d_in[10];
  const float* b2    = (const float*)d_in[11];
  float* out = (float*)d_out;

  const int N = in_sizes[0] / 128;   // 50000 nodes (= 3125 * 16)
  const int E = in_sizes[1] / 2;     // 800000 edges
  const int* src = ei;
  const int* dst = ei + E;

  // Workspace carve-up (floats)
  float* ws = (float*)d_ws;
  size_t off = 0;
  float*    XS   = ws + off; off += (size_t)N * 256;   // xs1, reused as hs2
  float*    Hb   = ws + off; off += (size_t)N * 256;   // out1 accum -> h
  float*    AS   = ws + off; off += (size_t)N * 8;
  float*    AD   = ws + off; off += (size_t)N * 8;
  unsigned* AMAX = (unsigned*)(ws + off); off += (size_t)N * 8;
  float*    DEN  = ws + off; off += (size_t)N * 8;
  float*    WA   = ws + off; off += 2048;
  float*    WD   = ws + off; off += 2048;

  const int TPB = 256;

  // ===================== Layer 1: heads=8, F=32, in=128 =====================
  // XS = x @ W1_src   [N,128]x[128,256]
  gemm_wmma_f32<256, 128><<<dim3(N / 16, 256 / 64), 32, 0, stream>>>(x, W1s, XS);
  // folded attention weights + projections (x @ W1_dst never materialized)
  make_wa<8, 32><<<(128 * 8 + TPB - 1) / TPB, TPB, 0, stream>>>(W1s, att1s, WA, 128);
  make_wa<8, 32><<<(128 * 8 + TPB - 1) / TPB, TPB, 0, stream>>>(W1d, att1d, WD, 128);
  proj_att<128, 8><<<((size_t)N * 8 + TPB - 1) / TPB, TPB, 0, stream>>>(x, WA, AS, N);
  proj_att<128, 8><<<((size_t)N * 8 + TPB - 1) / TPB, TPB, 0, stream>>>(x, WD, AD, N);

  hipMemsetAsync(AMAX, 0, (size_t)N * 8 * sizeof(unsigned), stream);
  hipMemsetAsync(DEN, 0, (size_t)N * 8 * sizeof(float), stream);
  hipMemsetAsync(Hb, 0, (size_t)N * 256 * sizeof(float), stream);

  int EH = E * 8;
  edge_max<8>  <<<(EH + TPB - 1) / TPB, TPB, 0, stream>>>(src, dst, AS, AD, AMAX, E);
  edge_denom<8><<<(EH + TPB - 1) / TPB, TPB, 0, stream>>>(src, dst, AS, AD, AMAX, DEN, E);
  edge_aggregate<8, 5><<<(E + 7) / 8, 256, 0, stream>>>(src, dst, AS, AD, AMAX, DEN,
                                                        XS, Hb, E);
  bias_relu<<<((size_t)N * 256 + TPB - 1) / TPB, TPB, 0, stream>>>(Hb, b1, N * 256, 256);

  // ===================== Layer 2: heads=1, F=128, in=256 ====================
  // HS2 = h @ W2_src  [N,256]x[256,128]  (reuses XS buffer)
  gemm_wmma_f32<128, 256><<<dim3(N / 16, 128 / 64), 32, 0, stream>>>(Hb, W2s, XS);
  make_wa<1, 128><<<(256 + TPB - 1) / TPB, TPB, 0, stream>>>(W2s, att2s, WA, 256);
  make_wa<1, 128><<<(256 + TPB - 1) / TPB, TPB, 0, stream>>>(W2d, att2d, WD, 256);
  proj_att<256, 1><<<(N + TPB - 1) / TPB, TPB, 0, stream>>>(Hb, WA, AS, N);
  proj_att<256, 1><<<(N + TPB - 1) / TPB, TPB, 0, stream>>>(Hb, WD, AD, N);

  hipMemsetAsync(AMAX, 0, (size_t)N * sizeof(unsigned), stream);
  hipMemsetAsync(DEN, 0, (size_t)N * sizeof(float), stream);
  hipMemsetAsync(out, 0, (size_t)N * 128 * sizeof(float), stream);

  edge_max<1>  <<<(E + TPB - 1) / TPB, TPB, 0, stream>>>(src, dst, AS, AD, AMAX, E);
  edge_denom<1><<<(E + TPB - 1) / TPB, TPB, 0, stream>>>(src, dst, AS, AD, AMAX, DEN, E);
  edge_aggregate<1, 7><<<(E + 7) / 8, 256, 0, stream>>>(src, dst, AS, AD, AMAX, DEN,
                                                        XS, out, E);
  // mean over 1 head == identity; add bias
  bias_add<<<((size_t)N * 128 + TPB - 1) / TPB, TPB, 0, stream>>>(out, b2, N * 128, 128);
}